// DOTA_mix_adaptive_36747740184585
// MI455X (gfx1250) — compile-verified
//
#include <hip/hip_runtime.h>
#include <hip/hip_bf16.h>
#include <math.h>

typedef __attribute__((ext_vector_type(2))) float v2f;
typedef __attribute__((ext_vector_type(8))) float v8f;

#define Bn   4096
#define Kc   1000
#define Mm   8
#define Dd   512
#define KM   (Kc * Mm)     // 8000
#define KDIM (2 * Dd)      // 1024 (concatenated [inv ; mu*inv])

// ---------------------------------------------------------------------------
// Prep: per (k,m) column build B-matrix column [inv ; mu*inv] (depth 1024),
// colconst = sum(mu^2*inv) + sum(log var), lpi = log(pi) or -1e30 if masked.
// Masked modes have pi_pad == 0 exactly, so the bool mask input is not needed.
// ---------------------------------------------------------------------------
__global__ __launch_bounds__(256) void dota_prep(
        const float* __restrict__ var_pad,
        const float* __restrict__ mu_pad,
        const float* __restrict__ pi_pad,
        float* __restrict__ Bmat,
        float* __restrict__ colconst,
        float* __restrict__ lpi) {
    const int km  = blockIdx.x;
    const int tid = threadIdx.x;
    const float* vp = var_pad + (size_t)km * Dd;
    const float* mp = mu_pad  + (size_t)km * Dd;
    float* bc = Bmat + (size_t)km * KDIM;

    float partial = 0.0f;
    for (int d = tid; d < Dd; d += 256) {
        float v   = vp[d] + 1e-3f;          // EPS_REG
        v         = fmaxf(v, 1e-8f);        // clip
        float inv = 1.0f / v;
        float mu  = mp[d];
        bc[d]      = inv;                   // pairs with x^2
        bc[Dd + d] = mu * inv;              // pairs with -2x
        partial += mu * mu * inv + __logf(v);
    }
    // wave32 butterfly + cross-wave LDS reduce
    for (int m = 1; m < 32; m <<= 1) partial += __shfl_xor(partial, m, 32);
    __shared__ float red[8];
    if ((tid & 31) == 0) red[tid >> 5] = partial;
    __syncthreads();
    if (tid == 0) {
        float tot = 0.0f;
        #pragma unroll
        for (int w = 0; w < 8; ++w) tot += red[w];
        colconst[km] = tot;
        float p = pi_pad[km];
        lpi[km] = (p > 0.0f) ? __logf(p + 1e-10f) : -1.0e30f;
    }
}

// ---------------------------------------------------------------------------
// Fused GEMM (f32 WMMA 16x16x4) + masked softmax over M=8 + gamma scaling.
// Block = 256 threads = 8 waves. Block tile: 128 rows x 16 cols.
// LDS holds the 16-column B tile (16 x 1024 f32 = 64 KB) in a float2-chunk
// layout: element (col,k) -> ldsB[(k>>1)*32 + col*2 + (k&1)], which makes the
// per-WMMA ds_load_b64 bank-conflict-free across the wave.
// ---------------------------------------------------------------------------
__global__ __launch_bounds__(256) void dota_maha_softmax(
        const float* __restrict__ x,
        const float* __restrict__ gamma_class,
        const float* __restrict__ Bmat,
        const float* __restrict__ colconst,
        const float* __restrict__ lpi,
        float* __restrict__ out) {
    __shared__ float ldsB[16 * KDIM];      // 64 KB

    const int tid     = threadIdx.x;
    const int colbase = blockIdx.x * 16;
    const int rowblk  = blockIdx.y * 128;

    // -------- stage 16 B-matrix columns into LDS (64 floats per thread) ----
    {
        const int col = tid >> 4;                  // 0..15
        const int k0  = (tid & 15) * 64;           // 64-deep slice
        const float* src = Bmat + (size_t)(colbase + col) * KDIM + k0;
        #pragma unroll
        for (int i = 0; i < 64; i += 4) {
            float4 q = *(const float4*)(src + i);
            const int k = k0 + i;                  // k is even
            float* dst = ldsB + (k >> 1) * 32 + col * 2;
            dst[0]  = q.x;                         // (k,   chunk j=0)
            dst[1]  = q.y;                         // (k+1, chunk j=1)
            dst[32] = q.z;                         // (k+2)
            dst[33] = q.w;                         // (k+3)
        }
    }
    __syncthreads();

    const int wave  = tid >> 5;
    const int lane  = tid & 31;
    const int lhalf = lane >> 4;                   // 0 or 1
    const int l16   = lane & 15;
    const int rowbase = rowblk + wave * 16;
    const int r       = rowbase + l16;             // this lane's A row

    // A fragment source: lane holds A[row][2*lhalf + {0,1}] per k-step
    const float* xrow = x + (size_t)r * Dd + 2 * lhalf;
    // B fragment source in LDS: chunk (kk/2 + lhalf), col l16
    const float* bls  = ldsB + lhalf * 32 + l16 * 2;

    v8f acc = {};
    #pragma unroll 8
    for (int kk = 0; kk < Dd; kk += 4) {
        v2f xa = *(const v2f*)(xrow + kk);
        // phase-1 A: x^2  against  inv
        v2f a1; a1.x = xa.x * xa.x;   a1.y = xa.y * xa.y;
        v2f b1 = *(const v2f*)(bls + kk * 16);
        acc = __builtin_amdgcn_wmma_f32_16x16x4_f32(
                  false, a1, false, b1, (short)0, acc, false, false);
        // phase-2 A: -2x  against  mu*inv  (second half of K-depth)
        v2f a2; a2.x = -2.0f * xa.x;  a2.y = -2.0f * xa.y;
        v2f b2 = *(const v2f*)(bls + (Dd + kk) * 16);
        acc = __builtin_amdgcn_wmma_f32_16x16x4_f32(
                  false, a2, false, b2, (short)0, acc, false, false);
    }

    // -------- epilogue: log-lik, masked softmax over M=8, gamma scale ------
    // C layout: VGPR v, lane -> row = rowbase + v + 8*lhalf, col = colbase+l16
    const int   col = colbase + l16;
    const int   kc  = col >> 3;                    // class index (M=8)
    const float cc  = colconst[col];
    const float lp  = lpi[col];

    #pragma unroll
    for (int v = 0; v < 8; ++v) {
        float t = fmaf(-0.5f, acc[v] + cc, lp);    // log_lik + log_pi
        // softmax over the 8 modes = butterfly over 8-lane groups
        float mx = t;
        mx = fmaxf(mx, __shfl_xor(mx, 1, 32));
        mx = fmaxf(mx, __shfl_xor(mx, 2, 32));
        mx = fmaxf(mx, __shfl_xor(mx, 4, 32));
        float e = __expf(t - mx);
        float s = e;
        s += __shfl_xor(s, 1, 32);
        s += __shfl_xor(s, 2, 32);
        s += __shfl_xor(s, 4, 32);
        const int b = rowbase + v + 8 * lhalf;
        const float g = gamma_class[(size_t)b * Kc + kc];
        out[(size_t)b * KM + col] = g * (e / s);
    }
}

// ---------------------------------------------------------------------------
extern "C" void kernel_launch(void* const* d_in, const int* in_sizes, int n_in,
                              void* d_out, int out_size, void* d_ws, size_t ws_size,
                              hipStream_t stream) {
    const float* x           = (const float*)d_in[0];
    const float* gamma_class = (const float*)d_in[1];
    const float* mu_pad      = (const float*)d_in[2];
    const float* var_pad     = (const float*)d_in[3];
    const float* pi_pad      = (const float*)d_in[4];
    // d_in[5] (mask) unused: pi_pad == 0 encodes masked modes exactly.

    float* Bmat     = (float*)d_ws;                       // 8000*1024 f32 = 32.77 MB
    float* colconst = Bmat + (size_t)KM * KDIM;           // 8000 f32
    float* lpiv     = colconst + KM;                      // 8000 f32

    dota_prep<<<dim3(KM), dim3(256), 0, stream>>>(
        var_pad, mu_pad, pi_pad, Bmat, colconst, lpiv);

    dim3 grid(KM / 16, Bn / 128);                         // 500 x 32 blocks
    dota_maha_softmax<<<grid, dim3(256), 0, stream>>>(
        x, gamma_class, Bmat, colconst, lpiv, (float*)d_out);
}